// BiMultiGCN_59854664237725
// MI455X (gfx1250) — compile-verified
//
#include <hip/hip_runtime.h>
#include <stdint.h>

// ---------------------------------------------------------------------------
// BiMultiGCN on MI455X (gfx1250).
// Pull (CSR-gather): one wave32 per destination row (lane = one float4 of the
// 128-float row), register accumulation, single store -- no f32 atomics.
// CSR entries packed 8B {src,int; w,float} -> one b64 load per edge.
// CSR build streams edge arrays through LDS with CDNA5 async-to-LDS
// (global_load_async_to_lds_b128 + s_wait_asynccnt), double buffered.
// Working set (~110 MB) fits the 192 MB L2, so the layer loop runs at L2 BW.
// ---------------------------------------------------------------------------

#define HID 128
#define TILE 256          // edges per LDS tile in build kernels
#define NBLK_EDGE 2048

__device__ __forceinline__ void async_load16(uint32_t lds_addr, uint64_t gaddr) {
  asm volatile("global_load_async_to_lds_b128 %0, %1, off"
               :: "v"(lds_addr), "v"(gaddr)
               : "memory");
}

__device__ __forceinline__ void wait_async0() {
  asm volatile("s_wait_asynccnt 0" ::: "memory");
}

__device__ __forceinline__ int atomic_add_i32(int* p, int v) {
  return __hip_atomic_fetch_add(p, v, __ATOMIC_RELAXED, __HIP_MEMORY_SCOPE_AGENT);
}

// ---------------------------------------------------------------------------

__global__ void fill_zero_int_kernel(int* __restrict__ p, size_t n) {
  size_t i = (size_t)blockIdx.x * blockDim.x + threadIdx.x;
  const size_t stride = (size_t)gridDim.x * blockDim.x;
  for (; i < n; i += stride) p[i] = 0;
}

__global__ void count_kernel(const int* __restrict__ dst, int* __restrict__ cnt, int E) {
  int i = blockIdx.x * blockDim.x + threadIdx.x;
  const int stride = gridDim.x * blockDim.x;
  for (; i < E; i += stride)
    (void)atomic_add_i32(&cnt[dst[i]], 1);
}

// single-workgroup exclusive scan: off[0]=0, off[i+1]=sum(cnt[0..i])
__global__ void scan_kernel(const int* __restrict__ cnt, int* __restrict__ off, int n) {
  __shared__ int sh[1024];
  __shared__ int carry;
  if (threadIdx.x == 0) carry = 0;
  __syncthreads();
  for (int base = 0; base < n; base += 1024) {
    const int i = base + (int)threadIdx.x;
    int v = (i < n) ? cnt[i] : 0;
    sh[threadIdx.x] = v;
    __syncthreads();
    for (int d = 1; d < 1024; d <<= 1) {
      int t = (threadIdx.x >= (unsigned)d) ? sh[threadIdx.x - d] : 0;
      __syncthreads();
      sh[threadIdx.x] += t;
      __syncthreads();
    }
    if (i < n) off[i + 1] = carry + sh[threadIdx.x];
    __syncthreads();
    if (threadIdx.x == 1023) carry += sh[1023];
    __syncthreads();
  }
  if (threadIdx.x == 0) off[0] = 0;
}

// CSR fill (weighted graph): csr[j] = {src[e], w[e]*wscale}, j from cursor.
// Edge streams staged via async-to-LDS, double buffered.
__global__ void __launch_bounds__(256)
fill_csr_kernel(const int* __restrict__ src, const int* __restrict__ dst,
                const float* __restrict__ wgt, float wscale,
                int* __restrict__ cursor, int2* __restrict__ csr, int E) {
  __shared__ int   sh_src[2][TILE];
  __shared__ int   sh_dst[2][TILE];
  __shared__ float sh_w[2][TILE];
  if (E <= 0) return;
  const int tiles = (E + TILE - 1) / TILE;
  const size_t gmax = ((size_t)E * 4 - 16) & ~(size_t)15;
  const int tid = threadIdx.x;
  const int a = tid >> 6, l = tid & 63;

  auto stage = [&](int tt, int pb) {
    if (tid >= 192) return;   // 3 streams x 64 chunks of 16B
    size_t goff = (size_t)tt * (TILE * 4) + (size_t)l * 16;
    if (goff > gmax) goff = gmax;
    const char* gp; uint32_t lp;
    if (a == 0)      { gp = (const char*)src; lp = (uint32_t)(uintptr_t)(&sh_src[pb][0]); }
    else if (a == 1) { gp = (const char*)dst; lp = (uint32_t)(uintptr_t)(&sh_dst[pb][0]); }
    else             { gp = (const char*)wgt; lp = (uint32_t)(uintptr_t)(&sh_w[pb][0]); }
    async_load16(lp + (uint32_t)(l * 16), (uint64_t)(uintptr_t)gp + goff);
  };

  int t = blockIdx.x, parity = 0;
  if (t < tiles) stage(t, 0);
  for (; t < tiles; t += gridDim.x) {
    wait_async0();
    __syncthreads();
    const int tn = t + gridDim.x;
    if (tn < tiles) stage(tn, parity ^ 1);
    const int nvalid = min(TILE, E - t * TILE);
    if (tid < nvalid) {
      const int   s = sh_src[parity][tid];
      const int   d = sh_dst[parity][tid];
      const float w = sh_w[parity][tid] * wscale;
      const int j = atomic_add_i32(&cursor[d], 1);
      csr[j] = make_int2(s, __float_as_int(w));
    }
    parity ^= 1;
  }
}

// CSR fill for the bipartite graph; weight = rsqrt(deg_dst * deg_src) derived
// from the two offset arrays (degree == offset difference).
__global__ void __launch_bounds__(256)
fill_csr_bip_kernel(const int* __restrict__ src, const int* __restrict__ dst,
                    const int* __restrict__ off_dst, const int* __restrict__ off_src,
                    int* __restrict__ cursor, int2* __restrict__ csr, int E) {
  __shared__ int sh_src[2][TILE];
  __shared__ int sh_dst[2][TILE];
  if (E <= 0) return;
  const int tiles = (E + TILE - 1) / TILE;
  const size_t gmax = ((size_t)E * 4 - 16) & ~(size_t)15;
  const int tid = threadIdx.x;
  const int a = tid >> 6, l = tid & 63;

  auto stage = [&](int tt, int pb) {
    if (tid >= 128) return;   // 2 streams x 64 chunks
    size_t goff = (size_t)tt * (TILE * 4) + (size_t)l * 16;
    if (goff > gmax) goff = gmax;
    const char* gp; uint32_t lp;
    if (a == 0) { gp = (const char*)src; lp = (uint32_t)(uintptr_t)(&sh_src[pb][0]); }
    else        { gp = (const char*)dst; lp = (uint32_t)(uintptr_t)(&sh_dst[pb][0]); }
    async_load16(lp + (uint32_t)(l * 16), (uint64_t)(uintptr_t)gp + goff);
  };

  int t = blockIdx.x, parity = 0;
  if (t < tiles) stage(t, 0);
  for (; t < tiles; t += gridDim.x) {
    wait_async0();
    __syncthreads();
    const int tn = t + gridDim.x;
    if (tn < tiles) stage(tn, parity ^ 1);
    const int nvalid = min(TILE, E - t * TILE);
    if (tid < nvalid) {
      const int s = sh_src[parity][tid];
      const int d = sh_dst[parity][tid];
      const float dd = (float)(off_dst[d + 1] - off_dst[d]);
      const float ds = (float)(off_src[s + 1] - off_src[s]);
      const float w  = rsqrtf(dd * ds);
      const int j = atomic_add_i32(&cursor[d], 1);
      csr[j] = make_int2(s, __float_as_int(w));
    }
    parity ^= 1;
  }
}

// Pull aggregation: one wave per destination row.
//   outx[d] = (basev ? basev[d] : 0) + sum_j w_j * x[src_j]
//   if (outacc) outacc[d] += outx[d]           (fused output accumulation)
__global__ void __launch_bounds__(256)
gather_kernel(const float* __restrict__ x,
              const int* __restrict__ off,
              const int2* __restrict__ csr,
              const float* __restrict__ basev,
              float* __restrict__ outx,
              float* __restrict__ outacc,
              int nnodes) {
  const int lane = threadIdx.x & 31;
  int wave = (int)((blockIdx.x * blockDim.x + threadIdx.x) >> 5);
  const int nw = (int)((gridDim.x * blockDim.x) >> 5);
  for (int d = wave; d < nnodes; d += nw) {
    const int jb = off[d], je = off[d + 1];
    float4 acc = make_float4(0.f, 0.f, 0.f, 0.f);
    for (int j = jb; j < je; ++j) {
      const int2  e = csr[j];                 // one b64 per edge (uniform addr)
      const int   s = e.x;
      const float w = __int_as_float(e.y);
      if (j + 1 < je) {
        __builtin_prefetch(x + (size_t)csr[j + 1].x * HID + lane * 4, 0, 0);
      }
      const float4 v = ((const float4*)(x + (size_t)s * HID))[lane];
      acc.x += w * v.x; acc.y += w * v.y; acc.z += w * v.z; acc.w += w * v.w;
    }
    if (basev) {
      const float4 b = ((const float4*)(basev + (size_t)d * HID))[lane];
      acc.x += b.x; acc.y += b.y; acc.z += b.z; acc.w += b.w;
    }
    ((float4*)(outx + (size_t)d * HID))[lane] = acc;
    if (outacc) {
      float4* p = (float4*)(outacc + (size_t)d * HID) + lane;
      float4 o = *p;
      o.x += acc.x; o.y += acc.y; o.z += acc.z; o.w += acc.w;
      *p = o;
    }
  }
}

__global__ void scale_kernel(float* __restrict__ out, size_t n4, float s) {
  size_t i = (size_t)blockIdx.x * blockDim.x + threadIdx.x;
  const size_t stride = (size_t)gridDim.x * blockDim.x;
  float4* o = (float4*)out;
  for (; i < n4; i += stride) {
    float4 v = o[i];
    v.x *= s; v.y *= s; v.z *= s; v.w *= s;
    o[i] = v;
  }
}

// ---------------------------------------------------------------------------

extern "C" void kernel_launch(void* const* d_in, const int* in_sizes, int n_in,
                              void* d_out, int out_size, void* d_ws, size_t ws_size,
                              hipStream_t stream) {
  (void)n_in; (void)out_size; (void)ws_size;

  const float* user_emb = (const float*)d_in[0];
  const float* spot_emb = (const float*)d_in[1];
  const int*   us_user  = (const int*)d_in[2];
  const int*   us_spot  = (const int*)d_in[3];
  const int*   u_ei  = (const int*)d_in[4];   const float* u_w  = (const float*)d_in[5];
  const int*   ci_ei = (const int*)d_in[6];   const float* ci_w = (const float*)d_in[7];
  const int*   st_ei = (const int*)d_in[8];   const float* st_w = (const float*)d_in[9];
  const int*   ca_ei = (const int*)d_in[10];  const float* ca_w = (const float*)d_in[11];

  const int E = in_sizes[2];
  const int N = in_sizes[0] / HID;   // 27094 users
  const int M = in_sizes[1] / HID;   // 42852 spots

  // ---- workspace carve (256B aligned) ----
  char* wp = (char*)d_ws;
  auto carve = [&](size_t bytes) {
    void* p = (void*)wp;
    wp += (bytes + 255) & ~(size_t)255;
    return p;
  };
  float* spotA   = (float*)carve((size_t)M * HID * 4);
  float* spotB   = (float*)carve((size_t)M * HID * 4);
  float* userA   = (float*)carve((size_t)N * HID * 4);
  int*   off_rel = (int*)carve((size_t)(M + 1) * 4);
  int*   off_uu  = (int*)carve((size_t)(N + 1) * 4);
  int*   off_bu  = (int*)carve((size_t)(N + 1) * 4);
  int*   off_bs  = (int*)carve((size_t)(M + 1) * 4);
  int*   cnt     = (int*)carve((size_t)M * 4);         // reused for all builds
  int*   cursor  = (int*)carve((size_t)M * 4);         // reused for all fills
  int2*  rel_csr = (int2*)carve((size_t)3 * E * 8);
  int2*  uu_csr  = (int2*)carve((size_t)E * 8);
  int2*  bu_csr  = (int2*)carve((size_t)E * 8);
  int2*  bs_csr  = (int2*)carve((size_t)E * 8);

  const dim3 blk(256);
  const int tiles = (E + TILE - 1) / TILE;
  const dim3 egrid((unsigned)(tiles < NBLK_EDGE ? tiles : NBLK_EDGE));
  const dim3 grid2k(2048);
  const dim3 grid256(256);
  const unsigned gb_m = (unsigned)((M + 7) / 8);   // 8 waves per 256-thr block
  const unsigned gb_n = (unsigned)((N + 7) / 8);
  const float third = 1.0f / 3.0f;

  // ---- merged relation CSR (3E edges onto spots, weights/3) ----
  fill_zero_int_kernel<<<grid256, blk, 0, stream>>>(cnt, (size_t)M);
  count_kernel<<<grid2k, blk, 0, stream>>>(ca_ei + E, cnt, E);
  count_kernel<<<grid2k, blk, 0, stream>>>(ci_ei + E, cnt, E);
  count_kernel<<<grid2k, blk, 0, stream>>>(st_ei + E, cnt, E);
  scan_kernel<<<1, 1024, 0, stream>>>(cnt, off_rel, M);
  hipMemcpyAsync(cursor, off_rel, (size_t)M * 4, hipMemcpyDeviceToDevice, stream);
  fill_csr_kernel<<<egrid, blk, 0, stream>>>(ca_ei, ca_ei + E, ca_w, third, cursor, rel_csr, E);
  fill_csr_kernel<<<egrid, blk, 0, stream>>>(ci_ei, ci_ei + E, ci_w, third, cursor, rel_csr, E);
  fill_csr_kernel<<<egrid, blk, 0, stream>>>(st_ei, st_ei + E, st_w, third, cursor, rel_csr, E);

  // ---- user-user CSR ----
  fill_zero_int_kernel<<<grid256, blk, 0, stream>>>(cnt, (size_t)N);
  count_kernel<<<grid2k, blk, 0, stream>>>(u_ei + E, cnt, E);
  scan_kernel<<<1, 1024, 0, stream>>>(cnt, off_uu, N);
  hipMemcpyAsync(cursor, off_uu, (size_t)N * 4, hipMemcpyDeviceToDevice, stream);
  fill_csr_kernel<<<egrid, blk, 0, stream>>>(u_ei, u_ei + E, u_w, 1.0f, cursor, uu_csr, E);

  // ---- bipartite CSRs (both directions); degrees come from the offsets ----
  fill_zero_int_kernel<<<grid256, blk, 0, stream>>>(cnt, (size_t)N);
  count_kernel<<<grid2k, blk, 0, stream>>>(us_user, cnt, E);
  scan_kernel<<<1, 1024, 0, stream>>>(cnt, off_bu, N);
  fill_zero_int_kernel<<<grid256, blk, 0, stream>>>(cnt, (size_t)M);
  count_kernel<<<grid2k, blk, 0, stream>>>(us_spot, cnt, E);
  scan_kernel<<<1, 1024, 0, stream>>>(cnt, off_bs, M);
  hipMemcpyAsync(cursor, off_bu, (size_t)N * 4, hipMemcpyDeviceToDevice, stream);
  fill_csr_bip_kernel<<<egrid, blk, 0, stream>>>(us_spot, us_user, off_bu, off_bs,
                                                 cursor, bu_csr, E);
  hipMemcpyAsync(cursor, off_bs, (size_t)M * 4, hipMemcpyDeviceToDevice, stream);
  fill_csr_bip_kernel<<<egrid, blk, 0, stream>>>(us_user, us_spot, off_bs, off_bu,
                                                 cursor, bs_csr, E);

  // ---- initial features: spot_x = spot_emb + rel-conv; user_x = user_emb + uu-conv
  gather_kernel<<<gb_m, blk, 0, stream>>>(spot_emb, off_rel, rel_csr,
                                          spot_emb, spotA, (float*)nullptr, M);
  gather_kernel<<<gb_n, blk, 0, stream>>>(user_emb, off_uu, uu_csr,
                                          user_emb, userA, (float*)nullptr, N);

  // ---- outputs: d_out = [spot_out | user_out], accumulated in place ----
  float* out_spot = (float*)d_out;
  float* out_user = out_spot + (size_t)M * HID;
  hipMemcpyAsync(out_spot, spotA, (size_t)M * HID * 4, hipMemcpyDeviceToDevice, stream);
  hipMemcpyAsync(out_user, userA, (size_t)N * HID * 4, hipMemcpyDeviceToDevice, stream);

  // ---- 3 bipartite layers; new_spot first (reads user), then user in-place
  float* cs = spotA;
  float* ns = spotB;
  float* cu = userA;
  for (int layer = 0; layer < 3; ++layer) {
    gather_kernel<<<gb_m, blk, 0, stream>>>(cu, off_bs, bs_csr,
                                            (const float*)nullptr, ns, out_spot, M);
    gather_kernel<<<gb_n, blk, 0, stream>>>(cs, off_bu, bu_csr,
                                            (const float*)nullptr, cu, out_user, N);
    float* t = cs; cs = ns; ns = t;
  }

  // ---- / (NUM_LAYERS + 1) ----
  scale_kernel<<<grid2k, blk, 0, stream>>>((float*)d_out, (size_t)(M + N) * HID / 4, 0.25f);
}